// FocalLoss_45956150067577
// MI455X (gfx1250) — compile-verified
//
#include <hip/hip_runtime.h>
#include <math.h>

// ---------------------------------------------------------------------------
// RetinaNet focal loss for MI455X (gfx1250, wave32).
// Memory-bound: 325 MB streamed once -> ~14 us floor at 23.3 TB/s.
// Block reductions use V_WMMA_F32_16X16X4_F32 (B = ones -> row sums, chained
// through the C accumulator) so the matrix pipe does the reduction in full
// f32 precision.
// ---------------------------------------------------------------------------

typedef float f4  __attribute__((ext_vector_type(4)));
typedef float v2f __attribute__((ext_vector_type(2)));
typedef float v8f __attribute__((ext_vector_type(8)));

#define ALPHA_  0.25f
#define EPS_    1e-4f

__device__ __forceinline__ float flog(float x) { return __logf(x); }

// Sum 256 f32 values resident in LDS using 4 chained f32 WMMAs.
// A-matrix (16x4 f32) layout: lanes 0-15 hold M=0..15 with VGPR0=K(base),
// VGPR1=K(base+1); lanes 16-31 hold K base = 2.  B = all-ones 4x16.
// D[m][n] accumulates row sums; per-lane sum of the 8 D VGPRs gives
// sum(M=0..7) on lanes 0-15 and sum(M=8..15) on lanes 16-31; xor-16 shuffle
// completes the total (broadcast to every lane).
__device__ __forceinline__ float wmma_sum256(const float* __restrict__ s, int lane) {
    const int m  = lane & 15;
    const int kb = (lane >> 4) << 1;          // 0 or 2
    v8f acc = {0.f, 0.f, 0.f, 0.f, 0.f, 0.f, 0.f, 0.f};
    v2f ones; ones[0] = 1.0f; ones[1] = 1.0f;
#pragma unroll
    for (int ch = 0; ch < 4; ++ch) {
        v2f a;
        a[0] = s[ch * 64 + m * 4 + kb];       // bank-conflict-free (stride 4/2)
        a[1] = s[ch * 64 + m * 4 + kb + 1];
        acc = __builtin_amdgcn_wmma_f32_16x16x4_f32(
            /*neg_a=*/false, a, /*neg_b=*/false, ones,
            /*c_mod=*/(short)0, acc, /*reuse_a=*/false, /*reuse_b=*/false);
    }
    float t = acc[0] + acc[1] + acc[2] + acc[3] + acc[4] + acc[5] + acc[6] + acc[7];
    t += __shfl_xor(t, 16, 32);
    return t;
}

__global__ __launch_bounds__(256) void focal_main(
    const float* __restrict__ cls,   // [B,A,C]
    const float* __restrict__ reg,   // [B,A,4]
    const float* __restrict__ anc,   // [1,A,4]
    const float* __restrict__ ann,   // [B,M,5]
    float* __restrict__ ws,          // [B,3] : cls_sum, reg_sum, num_pos
    int A, int C, int M)
{
    __shared__ float sAnn[5 * 64];        // up to 64 annotations
    __shared__ float sPart[3][256];

    const int tid = threadIdx.x;
    const int b   = blockIdx.y;
    const int a   = blockIdx.x * 256 + tid;

    for (int i = tid; i < M * 5; i += 256) sAnn[i] = ann[(size_t)b * M * 5 + i];
    __syncthreads();

    float clsSum = 0.f, regSum = 0.f, posf = 0.f;

    if (a < A) {
        // ---- anchor geometry -------------------------------------------------
        f4 av = *(const f4*)(anc + (size_t)a * 4);
        const float aw  = av[2] - av[0];
        const float ah  = av[3] - av[1];
        const float acx = av[0] + 0.5f * aw;
        const float acy = av[1] + 0.5f * ah;
        const float aArea = aw * ah;

        // ---- IoU max / argmax over M annotations (LDS broadcast reads) ------
        float best = -2.0f;   // iou values are >= -1, so m=0 always wins first
        int   bm   = 0;
        for (int m = 0; m < M; ++m) {
            const float bx1 = sAnn[m * 5 + 0];
            const float by1 = sAnn[m * 5 + 1];
            const float bx2 = sAnn[m * 5 + 2];
            const float by2 = sAnn[m * 5 + 3];
            const float lab = sAnn[m * 5 + 4];
            float iw = fmaxf(fminf(av[2], bx2) - fmaxf(av[0], bx1), 0.f);
            float ih = fmaxf(fminf(av[3], by2) - fmaxf(av[1], by1), 0.f);
            float inter = iw * ih;
            float ua = fmaxf(aArea + (bx2 - bx1) * (by2 - by1) - inter, 1e-8f);
            float iou = inter / ua;
            iou = (lab != -1.0f) ? iou : -1.0f;   // padded anns never win
            if (iou > best) { best = iou; bm = m; }
        }
        const bool pos = (best >= 0.5f);
        const bool neg = (best <  0.4f);

        if (pos || neg) {
            // ---- classification focal loss ----------------------------------
            // neg anchor: every class has target 0 -> l0 = 0.75*p^2*(-log(1-p))
            // pos anchor: same, then fix up the one-hot class afterwards.
            const float* cbase = cls + ((size_t)b * A + a) * C;
            const f4* cp = (const f4*)cbase;
            const int n4 = C >> 2;
            for (int j = 0; j < n4; ++j) {
                f4 v = __builtin_nontemporal_load(cp + j);   // streamed once: NT
#pragma unroll
                for (int e = 0; e < 4; ++e) {
                    float p = fminf(fmaxf(v[e], EPS_), 1.f - EPS_);
                    clsSum += (1.f - ALPHA_) * p * p * (-flog(1.f - p));
                }
            }
            for (int c = n4 << 2; c < C; ++c) {              // generic tail
                float p = fminf(fmaxf(cbase[c], EPS_), 1.f - EPS_);
                clsSum += (1.f - ALPHA_) * p * p * (-flog(1.f - p));
            }

            if (pos) {
                posf = 1.f;
                // fix up target class (reload hits L0; bit-identical recompute)
                const int tc = (int)sAnn[bm * 5 + 4];
                float p = fminf(fmaxf(cbase[tc], EPS_), 1.f - EPS_);
                clsSum -= (1.f - ALPHA_) * p * p * (-flog(1.f - p));
                clsSum += ALPHA_ * (1.f - p) * (1.f - p) * (-flog(p));

                // ---- regression smooth-L1 (only positives touch regressions)
                const float gx1 = sAnn[bm * 5 + 0];
                const float gy1 = sAnn[bm * 5 + 1];
                const float gx2 = sAnn[bm * 5 + 2];
                const float gy2 = sAnn[bm * 5 + 3];
                f4 rv = __builtin_nontemporal_load(
                            (const f4*)(reg + ((size_t)b * A + a) * 4));
                const float gwr = gx2 - gx1, ghr = gy2 - gy1;
                const float gcx = gx1 + 0.5f * gwr, gcy = gy1 + 0.5f * ghr;
                const float gw = fmaxf(gwr, 1.f), gh = fmaxf(ghr, 1.f);
                const float t0 = ((gcx - acx) / aw) / 0.1f;
                const float t1 = ((gcy - acy) / ah) / 0.1f;
                const float t2 = flog(gw / aw) / 0.2f;
                const float t3 = flog(gh / ah) / 0.2f;
                const float d0 = fabsf(t0 - rv[0]);
                const float d1 = fabsf(t1 - rv[1]);
                const float d2 = fabsf(t2 - rv[2]);
                const float d3 = fabsf(t3 - rv[3]);
                const float thr = 1.0f / 9.0f, sh = 0.5f / 9.0f;
                regSum  = (d0 <= thr) ? 4.5f * d0 * d0 : d0 - sh;
                regSum += (d1 <= thr) ? 4.5f * d1 * d1 : d1 - sh;
                regSum += (d2 <= thr) ? 4.5f * d2 * d2 : d2 - sh;
                regSum += (d3 <= thr) ? 4.5f * d3 * d3 : d3 - sh;
            }
        }
    }

    // ---- block reduction via f32 WMMA (wave 0, EXEC all-ones) ---------------
    sPart[0][tid] = clsSum;
    sPart[1][tid] = regSum;
    sPart[2][tid] = posf;
    __syncthreads();

    if (tid < 32) {
#pragma unroll
        for (int q = 0; q < 3; ++q) {
            float tot = wmma_sum256(sPart[q], tid);
            if (tid == 0) atomicAdd(ws + b * 3 + q, tot);
        }
    }
}

__global__ void zero_ws(float* ws, int n) {
    int i = blockIdx.x * blockDim.x + threadIdx.x;
    if (i < n) ws[i] = 0.f;
}

__global__ void finalize_k(const float* __restrict__ ws, float* __restrict__ out, int B) {
    if (threadIdx.x == 0 && blockIdx.x == 0) {
        float cm = 0.f, rm = 0.f;
        for (int b = 0; b < B; ++b) {
            const float cs = ws[b * 3 + 0];
            const float rs = ws[b * 3 + 1];
            const float np = ws[b * 3 + 2];
            cm += cs / fmaxf(np, 1.0f);
            rm += (np > 0.0f) ? rs / fmaxf(np * 4.0f, 1.0f) : 0.0f;
        }
        out[0] = cm / (float)B;
        out[1] = rm / (float)B;
    }
}

extern "C" void kernel_launch(void* const* d_in, const int* in_sizes, int n_in,
                              void* d_out, int out_size, void* d_ws, size_t ws_size,
                              hipStream_t stream) {
    const float* cls = (const float*)d_in[0];  // [B,A,C]
    const float* reg = (const float*)d_in[1];  // [B,A,4]
    const float* anc = (const float*)d_in[2];  // [1,A,4]
    const float* ann = (const float*)d_in[3];  // [B,M,5]

    const int A = in_sizes[2] / 4;
    const int B = in_sizes[1] / (A * 4);
    const int C = in_sizes[0] / (A * B);
    const int M = in_sizes[3] / (B * 5);

    float* ws  = (float*)d_ws;
    float* out = (float*)d_out;

    zero_ws<<<1, 64, 0, stream>>>(ws, 3 * B);

    dim3 grid((A + 255) / 256, B);
    focal_main<<<grid, 256, 0, stream>>>(cls, reg, anc, ann, ws, A, C, M);

    finalize_k<<<1, 32, 0, stream>>>(ws, out, B);
}